// GATModelBasic_6700148982286
// MI455X (gfx1250) — compile-verified
//
#include <hip/hip_runtime.h>
#include <math.h>

// ---------------------------------------------------------------------------
// Two-layer GATConv for gfx1250 (MI455X).
//  - GEMMs via V_WMMA_F32_16X16X4_F32 (wave32, full fp32 precision).
//    One wave computes a 16 x (16*NT) C strip. Software-pipelined k-loop:
//    next k-step's A/B fragments are loaded into fresh registers before the
//    current step's NT WMMAs execute, so matrix ops hide load latency.
//    NCOLS is a compile-time template parameter so all B offsets are affine
//    constants (no runtime multiplies); unsigned offsets avoid sign-extend
//    chains and keep loads in saddr+voffset form.
//  - segment softmax via monotone-uint atomicMax + f32 atomicAdd
//  - scatter aggregation via f32 atomicAdd (L2-resident working set)
// ---------------------------------------------------------------------------

typedef float v2f __attribute__((ext_vector_type(2)));
typedef float v8f __attribute__((ext_vector_type(8)));

#define NEG_SLOPE 0.2f
#define EPS_F 1e-16f
// monotone(-inf): bits(-inf)=0xFF800000 (negative) -> ~u = 0x007FFFFF
#define MONO_NEG_INF 0x007FFFFFu

__device__ __forceinline__ unsigned f2mono(float f) {
    unsigned u = __float_as_uint(f);
    return (u & 0x80000000u) ? ~u : (u | 0x80000000u);
}
__device__ __forceinline__ float mono2f(unsigned u) {
    return (u & 0x80000000u) ? __uint_as_float(u & 0x7FFFFFFFu)
                             : __uint_as_float(~u);
}

// ---------------- GEMM: C[M x NCOLS] = A[M x 128] * B[128 x NCOLS] ----------
// One wave computes rows [wave*16, wave*16+16) x all NT 16-col tiles.
// K fixed at 128 (both layers). Loads are clamped, never predicated.
template <int NT, int NCOLS>
__global__ void gemm_wmma_f32(const float* __restrict__ A,
                              const float* __restrict__ B,
                              float* __restrict__ C,
                              int M) {
    const unsigned K = 128;
    unsigned wave = (blockIdx.x * (unsigned)blockDim.x + threadIdx.x) >> 5;
    unsigned lane = threadIdx.x & 31;
    unsigned mtiles = ((unsigned)M + 15u) >> 4;
    if (wave >= mtiles) return;

    unsigned half = lane >> 4;   // 0: K={0,1}/rows 0-7 ; 1: K={2,3}/rows 8-15
    unsigned l16  = lane & 15;

    // unsigned 32-bit element offsets from uniform bases (saddr+voffset form)
    unsigned arow = min(wave * 16u + l16, (unsigned)(M - 1));   // clamp
    unsigned aoff = arow * K + half * 2u;                       // 8B aligned
    unsigned bcol[NT];
#pragma unroll
    for (int t = 0; t < NT; ++t)
        bcol[t] = min((unsigned)(t * 16) + l16, (unsigned)(NCOLS - 1));
    const unsigned brow = half * 2u * NCOLS;   // this half's K-row pair

    v8f acc[NT];
#pragma unroll
    for (int t = 0; t < NT; ++t) acc[t] = (v8f){};

    // ---- prologue: load k-step 0 ----
    v2f a_cur = *(const v2f*)(A + aoff);
    v2f b_cur[NT];
#pragma unroll
    for (int t = 0; t < NT; ++t) {
        b_cur[t].x = B[brow + bcol[t]];
        b_cur[t].y = B[brow + NCOLS + bcol[t]];
    }

    for (unsigned kk = 0; kk < K; kk += 4) {
        // ---- issue next k-step's loads first (latency hidden by WMMAs) ----
        unsigned kn = (kk + 4 < K) ? (kk + 4) : kk;   // safe reload on last
        v2f a_nxt = *(const v2f*)(A + aoff + kn);
        unsigned bn = brow + kn * NCOLS;              // affine, no runtime mul
        v2f b_nxt[NT];
#pragma unroll
        for (int t = 0; t < NT; ++t) {
            b_nxt[t].x = B[bn + bcol[t]];
            b_nxt[t].y = B[bn + NCOLS + bcol[t]];
        }
        // ---- NT independent accumulator chains on current fragments ----
#pragma unroll
        for (int t = 0; t < NT; ++t)
            acc[t] = __builtin_amdgcn_wmma_f32_16x16x4_f32(
                false, a_cur, false, b_cur[t], (short)0, acc[t], false, false);
        a_cur = a_nxt;
#pragma unroll
        for (int t = 0; t < NT; ++t) b_cur[t] = b_nxt[t];
    }

    unsigned crow0 = wave * 16u + half * 8u;
#pragma unroll
    for (int t = 0; t < NT; ++t) {
        unsigned c0 = (unsigned)(t * 16) + l16;
        if ((int)c0 < NCOLS) {
#pragma unroll
            for (int v = 0; v < 8; ++v) {
                unsigned r = crow0 + (unsigned)v;
                if ((int)r < M) C[r * (unsigned)NCOLS + c0] = acc[t][v];
            }
        }
    }
}

// ---------------- per-node attention coefficients ---------------------------
// out_s[n,h] = sum_f h[n,h,f]*a_s[h,f] ; likewise out_d.
__global__ void attn_coef_kernel(const float* __restrict__ h,
                                 const float* __restrict__ a_s,
                                 const float* __restrict__ a_d,
                                 float* __restrict__ out_s,
                                 float* __restrict__ out_d,
                                 int N, int H, int F) {
    int t = blockIdx.x * blockDim.x + threadIdx.x;
    if (t >= N * H) return;
    int n = t / H, hh = t % H;
    const float* row = h + (size_t)n * H * F + (size_t)hh * F;
    float ss = 0.f, dd = 0.f;
    for (int f = 0; f < F; ++f) {
        float v = row[f];
        ss = fmaf(v, a_s[hh * F + f], ss);
        dd = fmaf(v, a_d[hh * F + f], dd);
    }
    out_s[t] = ss;
    out_d[t] = dd;
}

// ---------------- init: out = bias broadcast, m = -inf, s = 0 ---------------
__global__ void init_layer_kernel(float* __restrict__ o,
                                  unsigned* __restrict__ m,
                                  float* __restrict__ s,
                                  const float* __restrict__ bias,
                                  long long totalC, int C, long long totalH) {
    long long t = blockIdx.x * (long long)blockDim.x + threadIdx.x;
    if (t < totalC) o[t] = bias[t % C];
    if (t < totalH) { m[t] = MONO_NEG_INF; s[t] = 0.f; }
}

__device__ __forceinline__ void edge_endpoints(const int* ei, int E, int e,
                                               int& s, int& d) {
    if (e < E) { s = ei[e]; d = ei[E + e]; }
    else       { s = d = e - E; }             // self-loop
}

// ---------------- edge pass 1: segment max of leaky_relu(as+ad) over dst ----
__global__ void edge_max_kernel(const int* __restrict__ ei, int E, int nE,
                                const float* __restrict__ as,
                                const float* __restrict__ ad,
                                unsigned* __restrict__ m, int H) {
    int t = blockIdx.x * blockDim.x + threadIdx.x;
    if (t >= nE * H) return;
    int e = t / H, hh = t % H;
    int s, d; edge_endpoints(ei, E, e, s, d);
    float el = as[(size_t)s * H + hh] + ad[(size_t)d * H + hh];
    el = el >= 0.f ? el : NEG_SLOPE * el;
    atomicMax(&m[(size_t)d * H + hh], f2mono(el));
}

// ---------------- edge pass 2: segment sum of exp(e - m[dst]) ---------------
__global__ void edge_sum_kernel(const int* __restrict__ ei, int E, int nE,
                                const float* __restrict__ as,
                                const float* __restrict__ ad,
                                const unsigned* __restrict__ m,
                                float* __restrict__ ssum, int H) {
    int t = blockIdx.x * blockDim.x + threadIdx.x;
    if (t >= nE * H) return;
    int e = t / H, hh = t % H;
    int s, d; edge_endpoints(ei, E, e, s, d);
    float el = as[(size_t)s * H + hh] + ad[(size_t)d * H + hh];
    el = el >= 0.f ? el : NEG_SLOPE * el;
    float p = __expf(el - mono2f(m[(size_t)d * H + hh]));
    atomicAdd(&ssum[(size_t)d * H + hh], p);
}

// ---------------- edge pass 3: out[dst] += alpha * h[src] -------------------
__global__ void edge_aggr_kernel(const int* __restrict__ ei, int E, int nE,
                                 const float* __restrict__ as,
                                 const float* __restrict__ ad,
                                 const unsigned* __restrict__ m,
                                 const float* __restrict__ ssum,
                                 const float* __restrict__ hbuf,
                                 float* __restrict__ obuf,
                                 int H, int F, long long total) {
    long long t = blockIdx.x * (long long)blockDim.x + threadIdx.x;
    if (t >= total) return;
    int HF = H * F;
    int e  = (int)(t / HF);
    int c  = (int)(t % HF);
    int hh = c / F;
    int s, d; edge_endpoints(ei, E, e, s, d);
    float el = as[(size_t)s * H + hh] + ad[(size_t)d * H + hh];
    el = el >= 0.f ? el : NEG_SLOPE * el;
    float p     = __expf(el - mono2f(m[(size_t)d * H + hh]));
    float alpha = p / (ssum[(size_t)d * H + hh] + EPS_F);
    float val   = alpha * hbuf[(size_t)s * HF + c];
    atomicAdd(&obuf[(size_t)d * HF + c], val);
}

// ---------------- relu in place ---------------------------------------------
__global__ void relu_kernel(float* __restrict__ x, long long total) {
    long long t = blockIdx.x * (long long)blockDim.x + threadIdx.x;
    if (t < total) x[t] = fmaxf(x[t], 0.f);
}

// ---------------- log_softmax over 40 classes, one wave per row -------------
__global__ void log_softmax40_kernel(const float* __restrict__ in,
                                     float* __restrict__ out, int N) {
    int wave = (int)((blockIdx.x * (unsigned)blockDim.x + threadIdx.x) >> 5);
    int lane = threadIdx.x & 31;
    if (wave >= N) return;
    const float* row = in + (size_t)wave * 40;
    float v0 = row[lane];
    float v1 = (lane + 32) < 40 ? row[lane + 32] : -INFINITY;
    float mx = fmaxf(v0, v1);
#pragma unroll
    for (int o = 16; o > 0; o >>= 1) mx = fmaxf(mx, __shfl_xor(mx, o, 32));
    float sm = __expf(v0 - mx) + (((lane + 32) < 40) ? __expf(v1 - mx) : 0.f);
#pragma unroll
    for (int o = 16; o > 0; o >>= 1) sm += __shfl_xor(sm, o, 32);
    float lse = mx + __logf(sm);
    float* orow = out + (size_t)wave * 40;
    orow[lane] = v0 - lse;
    if (lane + 32 < 40) orow[lane + 32] = v1 - lse;
}

// ---------------------------------------------------------------------------
static inline int cdiv_ll(long long a, int b) { return (int)((a + b - 1) / b); }

extern "C" void kernel_launch(void* const* d_in, const int* in_sizes, int n_in,
                              void* d_out, int out_size, void* d_ws, size_t ws_size,
                              hipStream_t stream) {
    const float* x      = (const float*)d_in[0];
    const int*   ei     = (const int*)d_in[1];
    const float* W1     = (const float*)d_in[2];
    const float* a_src1 = (const float*)d_in[3];
    const float* a_dst1 = (const float*)d_in[4];
    const float* b1     = (const float*)d_in[5];
    const float* W2     = (const float*)d_in[6];
    const float* a_src2 = (const float*)d_in[7];
    const float* a_dst2 = (const float*)d_in[8];
    const float* b2     = (const float*)d_in[9];
    float* out = (float*)d_out;

    const int IN = 128, H1 = 8, HID = 16, C1 = H1 * HID /*128*/, OUT = 40;
    int N  = in_sizes[0] / IN;
    int E  = in_sizes[1] / 2;
    int nE = E + N;

    // ---- workspace carve (all 4-byte elements) ----
    float* p = (float*)d_ws;
    float*    h1  = p; p += (size_t)N * C1;   // x @ W1
    float*    as1 = p; p += (size_t)N * H1;
    float*    ad1 = p; p += (size_t)N * H1;
    unsigned* m1  = (unsigned*)p; p += (size_t)N * H1;
    float*    s1  = p; p += (size_t)N * H1;
    float*    o1  = p; p += (size_t)N * C1;   // layer-1 output (post relu)
    float*    z2  = p; p += (size_t)N * OUT;  // o1 @ W2
    float*    as2 = p; p += (size_t)N;
    float*    ad2 = p; p += (size_t)N;
    unsigned* m2  = (unsigned*)p; p += (size_t)N;
    float*    s2  = p; p += (size_t)N;
    float*    o2  = p; p += (size_t)N * OUT;  // layer-2 output (pre log_softmax)

    const int BLK = 256;
    int mtiles = (N + 15) / 16;

    // ===================== Layer 1 (H=8, F=16) =====================
    // GEMM1: h1 = x @ W1 ; 1 wave per 16-row strip, NT=8 covers all 128 cols
    gemm_wmma_f32<8, 128><<<cdiv_ll((long long)mtiles * 32, BLK), BLK, 0, stream>>>(
        x, W1, h1, N);
    attn_coef_kernel<<<cdiv_ll((long long)N * H1, BLK), BLK, 0, stream>>>(
        h1, a_src1, a_dst1, as1, ad1, N, H1, HID);
    init_layer_kernel<<<cdiv_ll((long long)N * C1, BLK), BLK, 0, stream>>>(
        o1, m1, s1, b1, (long long)N * C1, C1, (long long)N * H1);
    edge_max_kernel<<<cdiv_ll((long long)nE * H1, BLK), BLK, 0, stream>>>(
        ei, E, nE, as1, ad1, m1, H1);
    edge_sum_kernel<<<cdiv_ll((long long)nE * H1, BLK), BLK, 0, stream>>>(
        ei, E, nE, as1, ad1, m1, s1, H1);
    {
        long long total = (long long)nE * H1 * HID;
        edge_aggr_kernel<<<cdiv_ll(total, BLK), BLK, 0, stream>>>(
            ei, E, nE, as1, ad1, m1, s1, h1, o1, H1, HID, total);
    }
    relu_kernel<<<cdiv_ll((long long)N * C1, BLK), BLK, 0, stream>>>(
        o1, (long long)N * C1);

    // ===================== Layer 2 (H=1, F=40) =====================
    // GEMM2: z2 = o1 @ W2 ; NT=3 covers 48 cols, stores masked at 40
    gemm_wmma_f32<3, 40><<<cdiv_ll((long long)mtiles * 32, BLK), BLK, 0, stream>>>(
        o1, W2, z2, N);
    attn_coef_kernel<<<cdiv_ll((long long)N, BLK), BLK, 0, stream>>>(
        z2, a_src2, a_dst2, as2, ad2, N, 1, OUT);
    init_layer_kernel<<<cdiv_ll((long long)N * OUT, BLK), BLK, 0, stream>>>(
        o2, m2, s2, b2, (long long)N * OUT, OUT, (long long)N);
    edge_max_kernel<<<cdiv_ll((long long)nE, BLK), BLK, 0, stream>>>(
        ei, E, nE, as2, ad2, m2, 1);
    edge_sum_kernel<<<cdiv_ll((long long)nE, BLK), BLK, 0, stream>>>(
        ei, E, nE, as2, ad2, m2, s2, 1);
    {
        long long total = (long long)nE * OUT;
        edge_aggr_kernel<<<cdiv_ll(total, BLK), BLK, 0, stream>>>(
            ei, E, nE, as2, ad2, m2, s2, z2, o2, 1, OUT, total);
    }

    // ===================== log_softmax =====================
    log_softmax40_kernel<<<cdiv_ll((long long)N * 32, BLK), BLK, 0, stream>>>(
        o2, out, N);
}